// GCN_50723563765864
// MI455X (gfx1250) — compile-verified
//
#include <hip/hip_runtime.h>

// ---------------------------------------------------------------------------
// GCN forward on MI455X (gfx1250, wave32).
// Dense GEMMs use V_WMMA_F32_16X16X4_F32 (native fp32 matrix op -> exact
// same numerics as the fp32 reference). Message passing uses L2-resident
// global_atomic_add_f32 scatter (full feature matrix = 51MB << 192MB L2).
// ---------------------------------------------------------------------------

typedef __attribute__((ext_vector_type(2))) float v2f;
typedef __attribute__((ext_vector_type(8))) float v8f;

#define NNODES 100000
#define NEDGES 600000
#define NGRAPH 1000
#define EMB    128

// ---------------- degree / norm precompute --------------------------------

__global__ void k_deg_init(float* __restrict__ deg) {
    int i = blockIdx.x * blockDim.x + threadIdx.x;
    if (i < NNODES) deg[i] = 1.0f;            // self-loop contribution
}

__global__ void k_deg_edges(const int* __restrict__ dst, float* __restrict__ deg) {
    int e = blockIdx.x * blockDim.x + threadIdx.x;
    if (e < NEDGES) atomicAdd(&deg[dst[e]], 1.0f);
}

__global__ void k_dinv(float* __restrict__ deg) {
    int i = blockIdx.x * blockDim.x + threadIdx.x;
    if (i < NNODES) {
        float d = deg[i];                      // always >= 1 (self-loop)
        deg[i] = rsqrtf(d);
    }
}

__global__ void k_norm(const int* __restrict__ src, const int* __restrict__ dst,
                       const float* __restrict__ dinv, float* __restrict__ norm) {
    int e = blockIdx.x * blockDim.x + threadIdx.x;
    if (e < NEDGES) norm[e] = dinv[src[e]] * dinv[dst[e]];
}

// ---------------- GEMM: [N,4] @ [4,128] -> [N,128] (single WMMA per tile) --

__global__ void k_gemm0_wmma(const float* __restrict__ X,
                             const float* __restrict__ W,   // [4,128]
                             float* __restrict__ Out) {
    const int wave = threadIdx.x >> 5;
    const int lane = threadIdx.x & 31;
    const int m    = lane & 15;
    const int kb   = (lane >> 4) << 1;        // 0 or 2 (A layout: lanes16-31 hold K=2,3)
    const int row0 = blockIdx.x * 16;
    const int n    = wave * 16 + (lane & 15);

    v2f a = *(const v2f*)(X + (row0 + m) * 4 + kb);
    v2f b;
    b[0] = W[(kb + 0) * EMB + n];
    b[1] = W[(kb + 1) * EMB + n];

    v8f c = {};
    c = __builtin_amdgcn_wmma_f32_16x16x4_f32(false, a, false, b,
                                              (short)0, c, false, false);

    const int rbase = (lane >> 4) * 8;        // C layout: lanes16-31 -> M = r+8
    #pragma unroll
    for (int r = 0; r < 8; ++r)
        Out[(row0 + rbase + r) * EMB + n] = c[r];
}

// ---------------- GEMM: [N,128] @ [128,128] -> [N,128] --------------------
// 256 threads = 8 waves. A-tile (16x128, 8KB) staged in LDS, shared by all
// waves; wave w produces columns [16w, 16w+16) via 32 unrolled f32 WMMAs.

__global__ void k_gemm128_wmma(const float* __restrict__ H,
                               const float* __restrict__ W,   // [128,128]
                               float* __restrict__ Out) {
    __shared__ float sA[16 * EMB];

    const int row0 = blockIdx.x * 16;

    // cooperative tile load: 2048 floats = 512 float4, 256 threads x 2
    {
        const float4* g4 = (const float4*)(H + (size_t)row0 * EMB);
        float4* s4 = (float4*)sA;
        s4[threadIdx.x]       = g4[threadIdx.x];
        s4[threadIdx.x + 256] = g4[threadIdx.x + 256];
    }
    __syncthreads();

    const int wave = threadIdx.x >> 5;
    const int lane = threadIdx.x & 31;
    const int m    = lane & 15;
    const int kb   = (lane >> 4) << 1;
    const int n    = wave * 16 + (lane & 15);

    v8f c = {};
    #pragma unroll
    for (int k0 = 0; k0 < EMB; k0 += 4) {
        v2f a = *(const v2f*)(sA + m * EMB + k0 + kb);
        v2f b;
        b[0] = W[(k0 + kb + 0) * EMB + n];
        b[1] = W[(k0 + kb + 1) * EMB + n];
        c = __builtin_amdgcn_wmma_f32_16x16x4_f32(false, a, false, b,
                                                  (short)0, c, false, false);
    }

    const int rbase = (lane >> 4) * 8;
    #pragma unroll
    for (int r = 0; r < 8; ++r)
        Out[(size_t)(row0 + rbase + r) * EMB + n] = c[r];
}

// ---------------- edge scatter: agg[dst] += lin[src] * norm ---------------
// 32 threads per edge, each handles one float4 (4 channels).

__global__ void k_scatter(const int* __restrict__ src, const int* __restrict__ dst,
                          const float* __restrict__ norm,
                          const float* __restrict__ lin, float* __restrict__ agg) {
    int tid = blockIdx.x * blockDim.x + threadIdx.x;
    int e = tid >> 5;
    int q = tid & 31;
    if (e < NEDGES) {
        int s = src[e], d = dst[e];
        float nm = norm[e];
        float4 v = *(const float4*)(lin + (size_t)s * EMB + q * 4);
        float* o = agg + (size_t)d * EMB + q * 4;
        atomicAdd(o + 0, v.x * nm);
        atomicAdd(o + 1, v.y * nm);
        atomicAdd(o + 2, v.z * nm);
        atomicAdd(o + 3, v.w * nm);
    }
}

// ---------------- epilogue: H = relu(agg + lin*dinv^2 + bias) (in-place) --

__global__ void k_finalize(float* __restrict__ H,          // holds agg in, features out
                           const float* __restrict__ lin,
                           const float* __restrict__ dinv,
                           const float* __restrict__ bias) {
    int idx = blockIdx.x * blockDim.x + threadIdx.x;   // N*EMB threads
    int i = idx >> 7;
    int c = idx & 127;
    float sn = dinv[i] * dinv[i];
    float v = H[idx] + lin[idx] * sn + bias[c];
    H[idx] = v > 0.0f ? v : 0.0f;
}

// ---------------- mean pool --------------------------------------------------

__global__ void k_pool(const int* __restrict__ batch, const float* __restrict__ H,
                       float* __restrict__ pooled) {
    int tid = blockIdx.x * blockDim.x + threadIdx.x;   // N*32 threads
    int i = tid >> 5;
    int q = tid & 31;
    if (i < NNODES) {
        int g = batch[i];
        float4 v = *(const float4*)(H + (size_t)i * EMB + q * 4);
        float* o = pooled + (size_t)g * EMB + q * 4;
        atomicAdd(o + 0, v.x);
        atomicAdd(o + 1, v.y);
        atomicAdd(o + 2, v.z);
        atomicAdd(o + 3, v.w);
    }
}

__global__ void k_count(const int* __restrict__ batch, float* __restrict__ cnt) {
    int i = blockIdx.x * blockDim.x + threadIdx.x;
    if (i < NNODES) atomicAdd(&cnt[batch[i]], 1.0f);
}

// ---------------- head MLP: relu([mean,xs] @ Wl1 + bl1) @ Wl2 + bl2 ---------

__global__ void k_mlp(const float* __restrict__ pooled, const float* __restrict__ cnt,
                      const float* __restrict__ xs,
                      const float* __restrict__ Wl1, const float* __restrict__ bl1,
                      const float* __restrict__ Wl2, const float* __restrict__ bl2,
                      float* __restrict__ out) {
    __shared__ float red[64];
    int g = blockIdx.x;
    int j = threadIdx.x;                               // 64 threads

    float inv = 1.0f / fmaxf(cnt[g], 1.0f);
    float s = bl1[j];
    #pragma unroll 4
    for (int c = 0; c < EMB; ++c)
        s += (pooled[(size_t)g * EMB + c] * inv) * Wl1[c * 64 + j];
    #pragma unroll
    for (int k = 0; k < 4; ++k)
        s += xs[g * 4 + k] * Wl1[(EMB + k) * 64 + j];
    s = fmaxf(s, 0.0f);
    red[j] = s * Wl2[j];
    __syncthreads();
    for (int off = 32; off > 0; off >>= 1) {
        if (j < off) red[j] += red[j + off];
        __syncthreads();
    }
    if (j == 0) out[g] = red[0] + bl2[0];
}

// ---------------------------------------------------------------------------

extern "C" void kernel_launch(void* const* d_in, const int* in_sizes, int n_in,
                              void* d_out, int out_size, void* d_ws, size_t ws_size,
                              hipStream_t stream) {
    const float* x     = (const float*)d_in[0];
    const int*   eidx  = (const int*)  d_in[1];
    const float* xs    = (const float*)d_in[2];
    const int*   batch = (const int*)  d_in[3];
    const float* W0  = (const float*)d_in[4];
    const float* b0  = (const float*)d_in[5];
    const float* Wls[3] = { (const float*)d_in[6], (const float*)d_in[8],
                            (const float*)d_in[10] };
    const float* bls[3] = { (const float*)d_in[7], (const float*)d_in[9],
                            (const float*)d_in[11] };
    const float* Wl1 = (const float*)d_in[12];
    const float* bl1 = (const float*)d_in[13];
    const float* Wl2 = (const float*)d_in[14];
    const float* bl2 = (const float*)d_in[15];
    float* out = (float*)d_out;

    const int* srcA = eidx;            // edge_index[0]
    const int* dstA = eidx + NEDGES;   // edge_index[1]

    // workspace carve (256B aligned slices)
    char* p = (char*)d_ws;
    auto carve = [&](size_t bytes) {
        char* r = p;
        p += (bytes + 255) & ~(size_t)255;
        return r;
    };
    float* dinv   = (float*)carve((size_t)NNODES * 4);
    float* norm   = (float*)carve((size_t)NEDGES * 4);
    float* bufH   = (float*)carve((size_t)NNODES * EMB * 4);  // features / agg
    float* bufL   = (float*)carve((size_t)NNODES * EMB * 4);  // linear out
    float* pooled = (float*)carve((size_t)NGRAPH * EMB * 4);
    float* cnt    = (float*)carve((size_t)NGRAPH * 4);
    (void)in_sizes; (void)n_in; (void)out_size; (void)ws_size;

    const int T = 256;

    // symmetric GCN normalization
    k_deg_init <<<(NNODES + T - 1) / T, T, 0, stream>>>(dinv);
    k_deg_edges<<<(NEDGES + T - 1) / T, T, 0, stream>>>(dstA, dinv);
    k_dinv     <<<(NNODES + T - 1) / T, T, 0, stream>>>(dinv);
    k_norm     <<<(NEDGES + T - 1) / T, T, 0, stream>>>(srcA, dstA, dinv, norm);

    const int gemmBlocks  = NNODES / 16;            // 6250 (exact)
    const int scatBlocks  = (NEDGES * 32) / T;      // 75000 (exact)
    const int finBlocks   = (NNODES * EMB) / T;     // 50000 (exact)

    // layer 0: lin = x @ W0 ; agg ; H = relu(agg + self + b0)
    k_gemm0_wmma<<<gemmBlocks, T, 0, stream>>>(x, W0, bufL);
    hipMemsetAsync(bufH, 0, (size_t)NNODES * EMB * 4, stream);
    k_scatter   <<<scatBlocks, T, 0, stream>>>(srcA, dstA, norm, bufL, bufH);
    k_finalize  <<<finBlocks, T, 0, stream>>>(bufH, bufL, dinv, b0);

    // layers 1..3
    for (int l = 0; l < 3; ++l) {
        k_gemm128_wmma<<<gemmBlocks, T, 0, stream>>>(bufH, Wls[l], bufL);
        hipMemsetAsync(bufH, 0, (size_t)NNODES * EMB * 4, stream);
        k_scatter     <<<scatBlocks, T, 0, stream>>>(srcA, dstA, norm, bufL, bufH);
        k_finalize    <<<finBlocks, T, 0, stream>>>(bufH, bufL, dinv, bls[l]);
    }

    // global mean pool + head MLP
    hipMemsetAsync(pooled, 0, (size_t)NGRAPH * EMB * 4, stream);
    hipMemsetAsync(cnt,    0, (size_t)NGRAPH * 4,       stream);
    k_pool <<<(NNODES * 32) / T, T, 0, stream>>>(batch, bufH, pooled);
    k_count<<<(NNODES + T - 1) / T, T, 0, stream>>>(batch, cnt);
    k_mlp  <<<NGRAPH, 64, 0, stream>>>(pooled, cnt, xs, Wl1, bl1, Wl2, bl2, out);
}